// ByteEmbedding_38826504356332
// MI455X (gfx1250) — compile-verified
//
#include <hip/hip_runtime.h>

#define B_       8
#define S_BYTES  8192
#define BYTE_DIM 128
#define EMB_DIM  1024
#define NUM_TOK  2048
#define SCALE_F  11.313708498984760f  /* sqrt(128) */

typedef __attribute__((ext_vector_type(2))) float v2f;
typedef __attribute__((ext_vector_type(4))) float v4f;
typedef __attribute__((ext_vector_type(8))) float v8f;

// ---------------------------------------------------------------------------
// Kernel 1: embedding gather + ragged segment-mean pooling.
// One wave32 per (batch, token). byte_groups is sorted per batch, so each wave
// binary-searches its segment bounds (deterministic, no float atomics), then
// each lane accumulates 4 of the 128 dims as a float4.
// ---------------------------------------------------------------------------
__global__ __launch_bounds__(256) void pool_kernel(const int* __restrict__ x,
                                                   const int* __restrict__ groups,
                                                   const float* __restrict__ emb,
                                                   float* __restrict__ grouped)
{
    const int wave = blockIdx.x * (blockDim.x >> 5) + (threadIdx.x >> 5);
    const int lane = threadIdx.x & 31;
    const int b = wave / NUM_TOK;
    const int t = wave % NUM_TOK;

    const int* g = groups + b * S_BYTES;

    // lower_bound(t)
    int lo = 0, hi = S_BYTES;
    while (lo < hi) { int mid = (lo + hi) >> 1; if (g[mid] < t) lo = mid + 1; else hi = mid; }
    const int start = lo;
    // upper_bound(t)
    hi = S_BYTES;
    while (lo < hi) { int mid = (lo + hi) >> 1; if (g[mid] <= t) lo = mid + 1; else hi = mid; }
    const int end = lo;

    const int* xb = x + b * S_BYTES;
    v4f acc = (v4f)0.0f;
    for (int s = start; s < end; ++s) {
        const int idx = xb[s];  // emb table is 192 KB -> L2/WGP$-resident gather
        acc += *(const v4f*)(emb + (size_t)idx * BYTE_DIM + lane * 4);
    }
    const int cnt = end - start;
    const float k = SCALE_F / (float)(cnt > 0 ? cnt : 1);
    *(v4f*)(grouped + (size_t)wave * BYTE_DIM + lane * 4) = acc * k;
}

// ---------------------------------------------------------------------------
// Kernel 2: out = grouped[16384,128] x W^T, W = out_proj_w [1024,128] row-major.
// Row n of W is the K-vector for output column n (contiguous -> float2 loads).
// Block: 8 waves, 64 output columns staged in LDS (32 KB). Each wave: one
// 16-row tile x 4 col-tiles, K=128 as 32 x V_WMMA_F32_16X16X4_F32.
//
// Fragment layouts per CDNA5 ISA 7.12.2:
//   A 16x4  (2 VGPR): lane l -> M=l%16; VGPR{0,1} = K{0,1} (lanes 0-15),
//                                      K{2,3} (lanes 16-31)
//   B 4x16  (2 VGPR): lane l -> N=l%16; same K striping as A
//   C/D 16x16 (8 VGPR): VGPR r -> M = r + 8*(l/16), N = l%16
// ---------------------------------------------------------------------------
__global__ __launch_bounds__(256) void gemm_kernel(const float* __restrict__ G,
                                                   const float* __restrict__ W,
                                                   float* __restrict__ out)
{
    __shared__ float sW[64 * BYTE_DIM];  // 32 KB of the 320 KB WGP LDS

    const int tid = threadIdx.x;
    const int n0  = blockIdx.y * 64;

    // cooperative 128B-per-lane staging of 64 W rows into LDS
    {
        const v4f* src = (const v4f*)(W + (size_t)n0 * BYTE_DIM);
        v4f*       dst = (v4f*)sW;
        for (int i = tid; i < 64 * BYTE_DIM / 4; i += 256) dst[i] = src[i];
    }
    __syncthreads();

    const int wave = tid >> 5;
    const int lane = tid & 31;
    const int half = lane >> 4;   // 0: K{0,1}, 1: K{2,3} within each K-step of 4
    const int l16  = lane & 15;   // M for A, N for B/C/D
    const int row0 = (blockIdx.x * 8 + wave) * 16;

    v8f c0 = (v8f)0.0f, c1 = (v8f)0.0f, c2 = (v8f)0.0f, c3 = (v8f)0.0f;

    const float* arow = G + (size_t)(row0 + l16) * BYTE_DIM + 2 * half;
    const float* brow = sW + l16 * BYTE_DIM + 2 * half;

    for (int kb = 0; kb < 32; ++kb) {
        const int k = kb * 4;
        const v2f a  = *(const v2f*)(arow + k);
        const v2f b0 = *(const v2f*)(brow + 0 * 16 * BYTE_DIM + k);
        const v2f b1 = *(const v2f*)(brow + 1 * 16 * BYTE_DIM + k);
        const v2f b2 = *(const v2f*)(brow + 2 * 16 * BYTE_DIM + k);
        const v2f b3 = *(const v2f*)(brow + 3 * 16 * BYTE_DIM + k);
        c0 = __builtin_amdgcn_wmma_f32_16x16x4_f32(false, a, false, b0, (short)0, c0, false, false);
        c1 = __builtin_amdgcn_wmma_f32_16x16x4_f32(false, a, false, b1, (short)0, c1, false, false);
        c2 = __builtin_amdgcn_wmma_f32_16x16x4_f32(false, a, false, b2, (short)0, c2, false, false);
        c3 = __builtin_amdgcn_wmma_f32_16x16x4_f32(false, a, false, b3, (short)0, c3, false, false);
    }

    // D layout: VGPR r holds M = r + 8*half, N = l16; lanes 0-15 / 16-31 each
    // cover a contiguous 64B run per store -> reasonable coalescing.
    float* orow = out + (size_t)row0 * EMB_DIM + n0 + l16;
#pragma unroll
    for (int r = 0; r < 8; ++r) {
        const size_t off = (size_t)(r + 8 * half) * EMB_DIM;
        orow[off +  0] = c0[r];
        orow[off + 16] = c1[r];
        orow[off + 32] = c2[r];
        orow[off + 48] = c3[r];
    }
}

extern "C" void kernel_launch(void* const* d_in, const int* in_sizes, int n_in,
                              void* d_out, int out_size, void* d_ws, size_t ws_size,
                              hipStream_t stream) {
    const int*   x      = (const int*)d_in[0];     // [8, 8192]
    const int*   groups = (const int*)d_in[1];     // [8, 8192] sorted per batch
    const float* emb    = (const float*)d_in[2];   // [384, 128]
    const float* W      = (const float*)d_in[3];   // [1024, 128]
    float*       out    = (float*)d_out;           // [8, 2048, 1024]
    float*       grouped = (float*)d_ws;           // [16384, 128] f32 = 8 MiB scratch

    // 16384 (batch,token) waves, 8 waves (256 thr) per block
    pool_kernel<<<dim3((B_ * NUM_TOK) / 8), 256, 0, stream>>>(x, groups, emb, grouped);

    // rows: 1024 row-tiles of 16 / 8 per block = 128; cols: 1024 / 64 = 16
    gemm_kernel<<<dim3(128, 16), 256, 0, stream>>>(grouped, W, out);
}